// MultiViTTokenMapper_738734375556
// MI455X (gfx1250) — compile-verified
//
#include <hip/hip_runtime.h>

// MultiViT token mapper for MI455X (gfx1250, wave32).
//
// Work split:
//  - gather_kernel: the bandwidth-bound part (~800 MB traffic @ 23.3 TB/s,
//    ~35-60 us). One wave per 3 KB output row; float4 (b128) coalesced copy,
//    non-temporal stores (write-once 403 MB output >> 192 MB L2) so the
//    patch_tokens read stream keeps the L2 (201 MB with ~2x reuse).
//  - fc_cls_kernel: tiny (8 x 768 x 1536) fp32 GEMM on the matrix pipe via
//    V_WMMA_F32_16X16X4_F32 (fp32 in / fp32 accumulate => reference
//    precision). Batch rows 8..15 of the 16-row M tile are padding: whatever
//    they accumulate only lands in D rows 8..15, which are never stored, so
//    no masking is needed — just a clamped (always-valid) address.

typedef float v2f __attribute__((ext_vector_type(2)));
typedef float v4f __attribute__((ext_vector_type(4)));
typedef float v8f __attribute__((ext_vector_type(8)));

constexpr int S_ = 2, B_ = 8, N_ = 4096, H_ = 768, T_ = 16384;
constexpr int K_ = S_ * H_;  // 1536

// ---------------------------------------------------------------------------
// Class-token FC: out[b,0,n] = sum_c cls[b,c] * W[n,c] + bias[n]
// One wave per 16-wide N tile (48 tiles). M=16 (batches 0..7 live, 8..15
// padding). K consumed 4 at a time by V_WMMA_F32_16X16X4_F32.
//
// Fragment layouts (ISA 7.12.2, wave32):
//   A 16x4 f32 : lanes 0-15 -> M=lane, K={0,1} in VGPR{0,1};
//                lanes 16-31 -> K={2,3}.  => one float2 load per lane.
//   B 4x16 f32 : lanes give N=lane%16 with the same K split.
//   D 16x16 f32: VGPR r, lanes 0-15 -> (M=r, N=lane).
// ---------------------------------------------------------------------------
__global__ __launch_bounds__(32) void fc_cls_kernel(
    const float* __restrict__ cls,   // class_tokens [S,B,1,H]
    const float* __restrict__ W,     // fc_weight    [H, S*H]
    const float* __restrict__ bias,  // fc_bias      [H]
    float* __restrict__ out)         // out          [B, 1+T, H]
{
  const int lane = threadIdx.x & 31;
  const int n0   = blockIdx.x * 16;       // N tile base (48 tiles over H=768)
  const int nl   = lane & 15;             // N within tile / M row for A
  const int klo  = (lane >> 4) << 1;      // K sub-offset per half-wave: 0 or 2
  const int mc   = (nl < B_) ? nl : 0;    // clamped batch row (8..15 -> 0)

  const size_t wrow = (size_t)(n0 + nl) * (size_t)K_;  // W row for this lane

  v8f acc = {};  // 16x16 f32 accumulator (8 VGPRs)

  for (int k = 0; k < K_; k += 4) {
    const int c0 = k + klo;               // this lane's first K index
    // cls[b, c] = class_tokens[s, b, 0, h] with s = c/H, h = c%H.
    // Each 4-aligned K group lies in one s (H % 4 == 0), so the float2
    // pair never straddles the image boundary.
    const int s  = (c0 >= H_) ? 1 : 0;
    const int hh = c0 - s * H_;

    const v2f av = *(const v2f*)(cls + ((size_t)(s * B_ + mc) * H_ + hh));
    const v2f bv = *(const v2f*)(W + wrow + c0);

    // D = A(16x4) x B(4x16) + C, fp32 throughout.
    acc = __builtin_amdgcn_wmma_f32_16x16x4_f32(
        /*neg_a=*/false, av, /*neg_b=*/false, bv,
        /*c_mod=*/(short)0, acc, /*reuse_a=*/false, /*reuse_b=*/false);
  }

  // D layout: lanes 0..15 hold (M=r, N=lane) in VGPR r. Only M=0..7 valid.
  if (lane < 16) {
    const float bb = bias[n0 + lane];
#pragma unroll
    for (int r = 0; r < B_; ++r) {
      out[(size_t)r * (size_t)(T_ + 1) * H_ + (size_t)(n0 + lane)] =
          acc[r] + bb;
    }
  }
}

// ---------------------------------------------------------------------------
// Row gather: out[b, 1+t, :] = patch_tokens[src_img[b,t], b, src_tok[b,t], :]
// One wave per 3 KB row; 6 x float4 per lane = fully coalesced b128 traffic.
// ---------------------------------------------------------------------------
__global__ __launch_bounds__(256) void gather_kernel(
    const float* __restrict__ patch,    // patch_tokens [S,B,N,H]
    const int* __restrict__ src_img,    // [B,T]
    const int* __restrict__ src_tok,    // [B,T]
    float* __restrict__ out)            // [B, 1+T, H]
{
  const int lane = threadIdx.x & 31;
  const int wave = threadIdx.x >> 5;
  const int r = blockIdx.x * 8 + wave;  // flat row id in [0, B*T)
  const int b = r >> 14;                // r / T   (T = 16384)
  const int t = r & (T_ - 1);           // r % T

  const int si = src_img[r];            // [B,T] row-major -> flat index r
  const int st = src_tok[r];

  const v4f* __restrict__ src =
      (const v4f*)(patch + ((size_t)(si * B_ + b) * (size_t)N_ + st) * H_);
  v4f* __restrict__ dst =
      (v4f*)(out + ((size_t)b * (size_t)(T_ + 1) + 1 + t) * (size_t)H_);

#pragma unroll
  for (int i = 0; i < (H_ / 4) / 32; ++i) {  // 192 float4 / 32 lanes = 6
    const v4f v = src[lane + i * 32];
    __builtin_nontemporal_store(v, dst + lane + i * 32);
  }
}

// ---------------------------------------------------------------------------
extern "C" void kernel_launch(void* const* d_in, const int* in_sizes, int n_in,
                              void* d_out, int out_size, void* d_ws,
                              size_t ws_size, hipStream_t stream) {
  const float* cls     = (const float*)d_in[0];  // class_tokens [S,B,1,H]
  const float* patch   = (const float*)d_in[1];  // patch_tokens [S,B,N,H]
  const int*   src_img = (const int*)d_in[2];    // [B,T]
  const int*   src_tok = (const int*)d_in[3];    // [B,T]
  const float* W       = (const float*)d_in[4];  // [H, S*H]
  const float* bias    = (const float*)d_in[5];  // [H]
  float* out = (float*)d_out;                    // [B, 1+T, H]

  fc_cls_kernel<<<H_ / 16, 32, 0, stream>>>(cls, W, bias, out);
  gather_kernel<<<(B_ * T_) / 8, 256, 0, stream>>>(patch, src_img, src_tok,
                                                   out);
}